// MusicGenerationLoss_46420006535650
// MI455X (gfx1250) — compile-verified
//
#include <hip/hip_runtime.h>
#include <math.h>

typedef float v2f __attribute__((ext_vector_type(2)));
typedef float v8f __attribute__((ext_vector_type(8)));

#define LSMOOTH 0.15f

__device__ __forceinline__ float warpSum(float v) {
#pragma unroll
  for (int o = 16; o > 0; o >>= 1) v += __shfl_down(v, o, 32);
  return v;
}

// 256-thread block sum, deterministic order, result broadcast to all threads.
__device__ float blockSum(float v) {
  __shared__ float buf[8];
  v = warpSum(v);
  const int lane = threadIdx.x & 31, wid = threadIdx.x >> 5;
  if (lane == 0) buf[wid] = v;
  __syncthreads();
  float r = 0.f;
  if (wid == 0) {
    r = (lane < 8) ? buf[lane] : 0.f;
#pragma unroll
    for (int o = 4; o > 0; o >>= 1) r += __shfl_down(r, o, 32);
    if (lane == 0) buf[0] = r;
  }
  __syncthreads();
  r = buf[0];
  __syncthreads();
  return r;
}

// ---------------- Kernel 1: token loss per row (online softmax) ----------------
__global__ void k_token_rows(const float* __restrict__ logits,
                             const int* __restrict__ targets,
                             float* __restrict__ row_loss,
                             float* __restrict__ row_mask, int V) {
  const int row = blockIdx.x;
  const float* p = logits + (size_t)row * (size_t)V;
  const int tid = threadIdx.x, nt = blockDim.x;

  float m = -INFINITY, s = 0.f, t = 0.f;
  if ((V & 3) == 0) {
    const float4* p4 = (const float4*)p;
    const int V4 = V >> 2;
    for (int i = tid; i < V4; i += nt) {
      float4 x4 = p4[i];
      float xs[4] = {x4.x, x4.y, x4.z, x4.w};
#pragma unroll
      for (int j = 0; j < 4; ++j) {
        float x = xs[j];
        t += x;
        if (x > m) { s = s * __expf(m - x) + 1.f; m = x; }
        else       { s += __expf(x - m); }
      }
    }
  } else {
    for (int i = tid; i < V; i += nt) {
      float x = p[i];
      t += x;
      if (x > m) { s = s * __expf(m - x) + 1.f; m = x; }
      else       { s += __expf(x - m); }
    }
  }

  // wave32 merge of (m, s) states + t sum
#pragma unroll
  for (int o = 16; o > 0; o >>= 1) {
    float mo = __shfl_down(m, o, 32);
    float so = __shfl_down(s, o, 32);
    float to = __shfl_down(t, o, 32);
    float M  = fmaxf(m, mo);
    float ea = (m  == M) ? 1.f : ((m  == -INFINITY) ? 0.f : __expf(m  - M));
    float eb = (mo == M) ? 1.f : ((mo == -INFINITY) ? 0.f : __expf(mo - M));
    s = s * ea + so * eb;
    m = M; t += to;
  }

  __shared__ float sm[8], ss[8], st[8];
  const int lane = tid & 31, wid = tid >> 5;
  if (lane == 0) { sm[wid] = m; ss[wid] = s; st[wid] = t; }
  __syncthreads();
  if (tid == 0) {
    float M = sm[0], S = ss[0], T = st[0];
    for (int w = 1; w < (nt >> 5); ++w) {
      float m2 = sm[w], s2 = ss[w], t2 = st[w];
      float Mn = fmaxf(M, m2);
      float ea = (M  == Mn) ? 1.f : ((M  == -INFINITY) ? 0.f : __expf(M  - Mn));
      float eb = (m2 == Mn) ? 1.f : ((m2 == -INFINITY) ? 0.f : __expf(m2 - Mn));
      S = S * ea + s2 * eb; M = Mn; T += t2;
    }
    float lse = M + __logf(S);
    int   tg  = targets[row];
    float xt  = p[tg];
    float per = (1.f - LSMOOTH) * (lse - xt) + LSMOOTH * (lse - T / (float)V);
    float msk = (tg != 0) ? 1.f : 0.f;
    row_loss[row] = per * msk;
    row_mask[row] = msk;
  }
}

// ---------------- Kernel 2: pitch-class indicator bitmasks ----------------
__global__ void k_pc_masks(const int* __restrict__ cp, const int* __restrict__ ct,
                           unsigned* __restrict__ mp, unsigned* __restrict__ mt, int T) {
  int r = blockIdx.x * blockDim.x + threadIdx.x;
  if (r >= T) return;
  unsigned a = 0u, b = 0u;
#pragma unroll
  for (int j = 0; j < 4; ++j) {
    a |= 1u << (cp[r * 4 + j] % 12);
    b |= 1u << (ct[r * 4 + j] % 12);
  }
  mp[r] = a; mt[r] = b;
}

// ---------------- Kernel 3: G = ind[2048x12] x templates[12x8] via WMMA f32 ----------------
// One 16-row tile per wave; K=12 done as 3 chained 16x16x4 f32 WMMAs (exact 0/1 math).
__global__ void k_prog_wmma(const unsigned* __restrict__ mp, float* __restrict__ G,
                            int nTiles) {
  const int tid   = threadIdx.x;
  const int lane  = tid & 31;
  const int wid   = tid >> 5;
  const int gwave = blockIdx.x * (blockDim.x >> 5) + wid;   // uniform per wave
  const int half  = lane >> 4;   // which half-wave
  const int n     = lane & 15;   // N column for B / C-D
  const int M     = lane & 15;   // M row for A

  // template pitch-class bitmasks: tm[k][j]
  const unsigned tm[2][4] = { {0x091u, 0x221u, 0x884u, 0x091u},
                              {0x089u, 0x121u, 0x484u, 0x089u} };

  // B operand (4x16, K x N), per ISA layout: lanes 0-15 hold K={c0,c0+1} in V0/V1,
  // lanes 16-31 hold K={c0+2,c0+3}. Column n = 4k + j; zero for n>=8.
  unsigned colmask = (n < 8) ? tm[n >> 2][n & 3] : 0u;
  v2f B[3];
#pragma unroll
  for (int c = 0; c < 3; ++c) {
    int pc0 = c * 4 + 2 * half;
    B[c].x = (float)((colmask >> (pc0 + 0)) & 1u);
    B[c].y = (float)((colmask >> (pc0 + 1)) & 1u);
  }

  if (gwave >= nTiles) return;   // wave-uniform; EXEC stays all-ones for WMMA

  const int t   = gwave;
  const int row = t * 16 + M;
  unsigned rm   = mp[row];

  v8f d = {0.f, 0.f, 0.f, 0.f, 0.f, 0.f, 0.f, 0.f};
#pragma unroll
  for (int c = 0; c < 3; ++c) {
    int pc0 = c * 4 + 2 * half;     // A 16x4 layout: same K split as B
    v2f a;
    a.x = (float)((rm >> (pc0 + 0)) & 1u);
    a.y = (float)((rm >> (pc0 + 1)) & 1u);
    d = __builtin_amdgcn_wmma_f32_16x16x4_f32(false, a, false, B[c],
                                              (short)0, d, false, false);
  }

  // C/D layout: VGPR p, lane -> D[M = p + 8*half][N = n]
  if (n < 8) {
#pragma unroll
    for (int p = 0; p < 8; ++p) {
      int r2 = t * 16 + p + 8 * half;
      G[r2 * 8 + n] = d[p];
    }
  }
}

// ---------------- Kernel 4: all small losses + final combine ----------------
__global__ void k_finalize(const float* __restrict__ row_loss,
                           const float* __restrict__ row_mask, int R,
                           const unsigned* __restrict__ mp,
                           const unsigned* __restrict__ mt, int T,
                           const float* __restrict__ G,
                           const float* __restrict__ pitch, int N,
                           const float* __restrict__ intervals,
                           const float* __restrict__ onset,
                           const float* __restrict__ dur,
                           float* __restrict__ out) {
  const int tid = threadIdx.x, nt = blockDim.x;

  // (a) token loss
  float ls = 0.f, ms = 0.f;
  for (int i = tid; i < R; i += nt) { ls += row_loss[i]; ms += row_mask[i]; }
  ls = blockSum(ls); ms = blockSum(ms);
  const float token = ls / ms;

  // (b) chord similarity (eps-normalized indicator dot, matches reference order)
  const float eps = 1e-6f;
  float simacc = 0.f;
  for (int r = tid; r < T; r += nt) {
    unsigned a = mp[r], b = mt[r];
    float sp = (float)__popc(a) + 12.f * eps;
    float st = (float)__popc(b) + 12.f * eps;
    float sim = 0.f;
    for (int c = 0; c < 12; ++c) {
      float pa = ((float)((a >> c) & 1u) + eps) / sp;
      float pb = ((float)((b >> c) & 1u) + eps) / st;
      sim += pa * pb;
    }
    simacc += fminf(fmaxf(sim, 0.f), 1.f);
  }
  simacc = blockSum(simacc);
  const float chord_sim = 1.f - simacc / (float)T;

  // (c) progression penalty from WMMA-produced G
  const int W = T - 3;
  float pen = 0.f;
  for (int w = tid; w < W; w += nt) {
    float best = 1e30f;
#pragma unroll
    for (int k = 0; k < 2; ++k) {
      float dsum = 0.f;
#pragma unroll
      for (int j = 0; j < 4; ++j) {
        int row = w + j;
        float inter = G[row * 8 + k * 4 + j];
        float uni   = (float)__popc(mp[row]) + 3.f - inter;  // |win| + |tmpl| - inter
        dsum += 1.f - inter / (uni + 1e-8f);
      }
      best = fminf(best, dsum * 0.25f);
    }
    pen += best;
  }
  pen = blockSum(pen) / (float)W;
  const float harmony = chord_sim + 0.5f * pen;

  // (d) contour
  float il = 0.f;
  for (int i = tid; i < N - 1; i += nt) il += fmaxf(fabsf(intervals[i]) - 12.f, 0.f);
  il = blockSum(il) / (float)(N - 1);
  float sm2 = 0.f;
  for (int i = tid; i < N - 2; i += nt)
    sm2 += fabsf(pitch[i + 2] - 2.f * pitch[i + 1] + pitch[i]);
  sm2 = blockSum(sm2) / (float)(N - 2);
  float psum = 0.f;
  for (int i = tid; i < N; i += nt) psum += pitch[i];
  psum = blockSum(psum);
  const float mean = psum / (float)N;
  float ssq = 0.f;
  for (int i = tid; i < N; i += nt) { float d = pitch[i] - mean; ssq += d * d; }
  ssq = blockSum(ssq);
  float sd = sqrtf(ssq / (float)(N - 1));   // ddof=1
  if (sd == 0.f) sd = 1.f;
  float arch = 0.f;
  for (int i = tid; i < N; i += nt) {
    float x = -1.f + 2.f * (float)i / (float)(N - 1);
    float z = (pitch[i] - mean) / sd + x * x;  // z - (-x^2)
    arch += z * z;
  }
  arch = blockSum(arch) / (float)N;
  const float contour = il + 0.3f * sm2 + 0.3f * arch;

  // (e) rhythm
  const float bw[4] = {1.f, 0.5f, 0.75f, 0.5f};
  float bacc = 0.f;
  for (int i = tid; i < N; i += nt) {
    int pos = (int)fmodf(onset[i], 4.0f);
    pos = pos < 0 ? 0 : (pos > 3 ? 3 : pos);
    bacc += bw[pos];
  }
  bacc = blockSum(bacc);
  const float beat = 1.f - bacc / (float)N;
  const float reps = 1e-7f;
  float racc = 0.f;
  for (int i = tid; i < N - 1; i += nt) racc += (dur[i + 1] + reps) / (dur[i] + reps);
  racc = blockSum(racc);
  const float rmean = racc / (float)(N - 1);
  float rvar = 0.f;
  for (int i = tid; i < N - 1; i += nt) {
    float d = (dur[i + 1] + reps) / (dur[i] + reps) - rmean;
    rvar += d * d;
  }
  rvar = blockSum(rvar) / (float)(N - 1);
  const float rhythm = beat + 0.5f * tanhf(rvar);

  if (tid == 0) out[0] = token + 0.1f * (harmony + rhythm + contour);
}

extern "C" void kernel_launch(void* const* d_in, const int* in_sizes, int n_in,
                              void* d_out, int out_size, void* d_ws, size_t ws_size,
                              hipStream_t stream) {
  (void)n_in; (void)out_size; (void)ws_size;
  const float* logits    = (const float*)d_in[0];
  const int*   targets   = (const int*)d_in[1];
  const int*   cp        = (const int*)d_in[2];
  const int*   ct        = (const int*)d_in[3];
  const float* pitch     = (const float*)d_in[4];
  const float* intervals = (const float*)d_in[5];
  const float* onset     = (const float*)d_in[6];
  const float* dur       = (const float*)d_in[7];

  const int R = in_sizes[1];                 // 4096 rows (B*S)
  const int V = in_sizes[0] / R;             // 32000 vocab
  const int T = in_sizes[2] / 4;             // 2048 chord steps
  const int N = in_sizes[4];                 // 8192 pitch samples

  float*    row_loss = (float*)d_ws;
  float*    row_mask = row_loss + R;
  unsigned* mp       = (unsigned*)(row_mask + R);
  unsigned* mt       = mp + T;
  float*    G        = (float*)(mt + T);     // T x 8

  k_token_rows<<<R, 256, 0, stream>>>(logits, targets, row_loss, row_mask, V);
  k_pc_masks<<<(T + 255) / 256, 256, 0, stream>>>(cp, ct, mp, mt, T);

  const int nTiles = T / 16;                 // 128
  const int wavesPerBlock = 8;               // 256 threads, wave32
  const int blocks = (nTiles + wavesPerBlock - 1) / wavesPerBlock;
  k_prog_wmma<<<blocks, 256, 0, stream>>>(mp, G, nTiles);

  k_finalize<<<1, 256, 0, stream>>>(row_loss, row_mask, R, mp, mt, T, G,
                                    pitch, N, intervals, onset, dur,
                                    (float*)d_out);
}